// forward_warp_21406117003336
// MI455X (gfx1250) — compile-verified
//
#include <hip/hip_runtime.h>

#define BB 8
#define CC 3
#define HH 720
#define WW 1280
#define HW (HH * WW)        // 921600
#define CHW (CC * HW)       // 2764800
#define NPIX (BB * HW)      // 7372800
#define NOUT (BB * CHW)     // 22118400

typedef float v2f __attribute__((ext_vector_type(2)));
typedef float v4f __attribute__((ext_vector_type(4)));

// ---------------------------------------------------------------------------
// Kernel 1: zero the output (harness poisons d_out; we must re-zero per call).
// float4 stores; NOUT = 22118400 = 5529600 float4, exactly 21600 blocks x 256.
// ---------------------------------------------------------------------------
__global__ void __launch_bounds__(256)
zero_out_kernel(float* __restrict__ out) {
    int i = blockIdx.x * 256 + threadIdx.x;        // float4 index
    v4f z = {0.f, 0.f, 0.f, 0.f};
    ((v4f*)out)[i] = z;
}

// ---------------------------------------------------------------------------
// Kernel 2: forward bilinear splat.
// grid = (HW/256, B); one thread per source pixel.
//   - flow load: coalesced 8B non-temporal (streamed once, keep out of L2)
//   - im0 loads: 3 coalesced 4B non-temporal loads (channel stride HW)
//   - scatter: up to 12 non-returning global_atomic_add_f32 (resolve in L2;
//     output array (88.5 MB) is L2-resident on MI455X's 192 MB L2)
// Invalid corners are skipped — identical to reference, which zeroes their
// weight before the scatter-add.
// ---------------------------------------------------------------------------
__global__ void __launch_bounds__(256)
forward_warp_kernel(const float* __restrict__ im0,
                    const float* __restrict__ flow,
                    float* __restrict__ out) {
    const int p = blockIdx.x * 256 + threadIdx.x;  // pixel index within image
    const int b = blockIdx.y;

    const int y = p / WW;
    const int x = p - y * WW;

    // flow layout [B,H,W,2] -> float2 per pixel, fully coalesced
    const v2f f = __builtin_nontemporal_load((const v2f*)flow + b * HW + p);

    const float fx = (float)x + f.x;
    const float fy = (float)y + f.y;

    const float x0f = floorf(fx);
    const float y0f = floorf(fy);
    const float wx1 = fx - x0f;
    const float wx0 = 1.0f - wx1;
    const float wy1 = fy - y0f;
    const float wy0 = 1.0f - wy1;

    const int x0 = (int)x0f;
    const int y0 = (int)y0f;
    const int x1 = x0 + 1;
    const int y1 = y0 + 1;

    // source pixel values, 3 channels (im0 layout [B,C,H,W])
    const float* src = im0 + (size_t)b * CHW + p;
    const float v0 = __builtin_nontemporal_load(src);
    const float v1 = __builtin_nontemporal_load(src + HW);
    const float v2 = __builtin_nontemporal_load(src + 2 * HW);

    float* ob = out + (size_t)b * CHW;

    const bool vx0 = (unsigned)x0 < (unsigned)WW;
    const bool vx1 = (unsigned)x1 < (unsigned)WW;
    const bool vy0 = (unsigned)y0 < (unsigned)HH;
    const bool vy1 = (unsigned)y1 < (unsigned)HH;

    // non-returning fp32 add atomics -> global_atomic_add_f32 (STOREcnt path)
    #define SPLAT(off, wc)                                    \
        do {                                                  \
            float* q = ob + (off);                            \
            unsafeAtomicAdd(q,           (wc) * v0);          \
            unsafeAtomicAdd(q + HW,      (wc) * v1);          \
            unsafeAtomicAdd(q + 2 * HW,  (wc) * v2);          \
        } while (0)

    if (vy0) {
        const int row0 = y0 * WW;
        if (vx0) SPLAT(row0 + x0, wx0 * wy0);
        if (vx1) SPLAT(row0 + x1, wx1 * wy0);
    }
    if (vy1) {
        const int row1 = y1 * WW;
        if (vx0) SPLAT(row1 + x0, wx0 * wy1);
        if (vx1) SPLAT(row1 + x1, wx1 * wy1);
    }
    #undef SPLAT
}

extern "C" void kernel_launch(void* const* d_in, const int* in_sizes, int n_in,
                              void* d_out, int out_size, void* d_ws, size_t ws_size,
                              hipStream_t stream) {
    const float* im0  = (const float*)d_in[0];   // [B,C,H,W] fp32
    const float* flow = (const float*)d_in[1];   // [B,H,W,2] fp32
    float* out = (float*)d_out;                  // [B,C,H,W] fp32

    // 1) zero the output (NOUT/4 float4 elements, 256 threads/block)
    {
        const int n4 = NOUT / 4;                 // 5529600
        dim3 grid(n4 / 256);                     // 21600, exact
        zero_out_kernel<<<grid, 256, 0, stream>>>(out);
    }

    // 2) splat
    {
        dim3 grid(HW / 256, BB);                 // 3600 x 8, exact
        forward_warp_kernel<<<grid, 256, 0, stream>>>(im0, flow, out);
    }
}